// CLAPP_layer_segmented_15659450761259
// MI455X (gfx1250) — compile-verified
//
#include <hip/hip_runtime.h>

#define N_IN 8192
#define N_H  8192
#define PI_F 3.14159265358979323846f

typedef __attribute__((ext_vector_type(2))) float v2f;
typedef __attribute__((ext_vector_type(4))) float v4f;
typedef __attribute__((ext_vector_type(8))) float v8f;

// ---------------------------------------------------------------------------
// Kernel A: cur = W @ inp via V_WMMA_F32_16X16X4_F32, then LIF epilogue.
// One 16-row tile of W per block; 8 waves split K (1024 each); LDS reduce.
// A-matrix layout (ISA 7.12.2, 32-bit 16x4): lanes 0-15 = rows, VGPR0/1 = K
// pair, lane-half selects K half. We permute K inside each 8-chunk (legal:
// same permutation applied to A and B), so each lane issues one b128 load of
// W and one b128 load of inp per 2 WMMAs.
// ---------------------------------------------------------------------------
__global__ __launch_bounds__(256) void gemv_lif_kernel(
    const float* __restrict__ W, const float* __restrict__ inp,
    const float* __restrict__ mem, const float* __restrict__ spk_trace,
    const float* __restrict__ prediction, const float* __restrict__ neg_trace,
    float* __restrict__ spk_out, float* __restrict__ st_out,
    float* __restrict__ g_pos, float* __restrict__ g_neg)
{
    __shared__ float red[8][32][8];                    // 8 KB of 320 KB WGP LDS
    const int tid  = threadIdx.x;
    const int wave = tid >> 5;
    const int lane = tid & 31;
    const int half = lane >> 4;
    const int r0   = blockIdx.x << 4;
    const int row  = r0 + (lane & 15);
    const float* wrow = W + (size_t)row * N_IN + half * 4;
    const float* xptr = inp + half * 4;
    const int kbeg = wave * (N_IN / 8);
    const int kend = kbeg + (N_IN / 8);

    v8f acc = {};
    for (int k0 = kbeg; k0 < kend; k0 += 8) {
        v4f a = *(const v4f*)(wrow + k0);              // global_load_b128
        v4f b = *(const v4f*)(xptr + k0);              // broadcast within half
        v2f a0 = {a.x, a.y}, a1 = {a.z, a.w};
        v2f b0 = {b.x, b.y}, b1 = {b.z, b.w};
        acc = __builtin_amdgcn_wmma_f32_16x16x4_f32(
            false, a0, false, b0, (short)0, acc, false, false);
        acc = __builtin_amdgcn_wmma_f32_16x16x4_f32(
            false, a1, false, b1, (short)0, acc, false, false);
    }

    #pragma unroll
    for (int v = 0; v < 8; ++v) red[wave][lane][v] = acc[v];
    __syncthreads();

    // All N-columns of D are identical by construction; lane 0 holds rows
    // M=0..7 (VGPR v -> M=v), lane 16 holds rows M=8..15.
    if (wave == 0 && (lane & 15) == 0) {
        #pragma unroll
        for (int v = 0; v < 8; ++v) {
            float cur = 0.f;
            #pragma unroll
            for (int w = 0; w < 8; ++w) cur += red[w][lane][v];
            const int r = r0 + half * 8 + v;
            const float mv    = mem[r];
            const float reset = mv > 1.0f ? 1.0f : 0.0f;   // THR = 1
            const float mn    = 0.9f * mv + cur - reset;   // BETA = 0.9
            const float spk   = mn > 1.0f ? 1.0f : 0.0f;
            const float st    = spk_trace[r] + spk * 0.1f;
            const float d     = PI_F * (mn - 1.0f);
            const float surr  = 1.0f / (PI_F * (1.0f + d * d));
            spk_out[r] = spk;
            st_out[r]  = st;
            g_pos[r]   = (prediction[r] - st) * surr;
            g_neg[r]   = (neg_trace[r]  - st) * surr;
        }
    }
}

// ---------------------------------------------------------------------------
// Kernel B: loss reductions, scalar output, and coef[i] = f_pos*g_pos[i]
// - f_neg*g_neg[i] with the data-dependent gates folded in. Single block.
// ---------------------------------------------------------------------------
__global__ __launch_bounds__(256) void loss_coef_kernel(
    const float* __restrict__ st, const float* __restrict__ prediction,
    const float* __restrict__ neg_trace,
    const float* __restrict__ g_pos, const float* __restrict__ g_neg,
    float* __restrict__ scalar_out, float* __restrict__ coef)
{
    __shared__ float sl[256], sc[256];
    __shared__ float flags[2];
    const int tid = threadIdx.x;
    float l = 0.f, c = 0.f;
    for (int i = tid; i < N_H; i += 256) {
        float d1 = st[i]        - prediction[i];
        float d2 = neg_trace[i] - prediction[i];
        l += d1 * d1;
        c += d2 * d2;
    }
    sl[tid] = l; sc[tid] = c;
    __syncthreads();
    for (int s = 128; s > 0; s >>= 1) {
        if (tid < s) { sl[tid] += sl[tid + s]; sc[tid] += sc[tid + s]; }
        __syncthreads();
    }
    if (tid == 0) {
        const float loss   = sl[0];
        const float loss_c = -sc[0];
        scalar_out[0] = 0.5f * (loss + loss_c);
        flags[0] = loss   >  20.0f  ? 1.0f : 0.0f;
        flags[1] = loss_c > -100.0f ? 1.0f : 0.0f;
    }
    __syncthreads();
    const float fp = flags[0], fn = flags[1];
    for (int i = tid; i < N_H; i += 256)
        coef[i] = fp * g_pos[i] - fn * g_neg[i];
}

// ---------------------------------------------------------------------------
// Kernel C: dW[i][j] = coef[i] * (inp_trace[j] + 0.1*inp[j]).
// 256 MB of write-once data -> nontemporal stores. dW starts at a 4B-aligned
// offset (16385 floats) into d_out, so use b32 stores (fully coalesced:
// 128B/wave/iteration). inp/inp_trace (64 KB) stay L2-resident.
// ---------------------------------------------------------------------------
__global__ __launch_bounds__(256) void outer_kernel(
    const float* __restrict__ inp, const float* __restrict__ inp_trace,
    const float* __restrict__ coef, float* __restrict__ dW)
{
    const int i = blockIdx.x;
    const float ci = coef[i];                          // wave-uniform
    float* out = dW + (size_t)i * N_IN;
    for (int j = threadIdx.x; j < N_IN; j += 256) {
        const float it = inp_trace[j] + 0.1f * inp[j];
        __builtin_nontemporal_store(ci * it, out + j);
    }
}

extern "C" void kernel_launch(void* const* d_in, const int* in_sizes, int n_in,
                              void* d_out, int out_size, void* d_ws, size_t ws_size,
                              hipStream_t stream)
{
    const float* inp        = (const float*)d_in[0];
    const float* mem        = (const float*)d_in[1];
    const float* spk_trace  = (const float*)d_in[2];
    const float* inp_trace  = (const float*)d_in[3];
    const float* prediction = (const float*)d_in[4];
    const float* neg_trace  = (const float*)d_in[5];
    const float* W          = (const float*)d_in[6];

    float* out        = (float*)d_out;
    float* spk_out    = out;                 // [8192]
    float* st_out     = out + N_H;           // [8192]
    float* scalar_out = out + 2 * N_H;       // [1]
    float* dW         = out + 2 * N_H + 1;   // [8192*8192]

    float* ws    = (float*)d_ws;
    float* g_pos = ws;                       // [8192]
    float* g_neg = ws + N_H;                 // [8192]
    float* coef  = ws + 2 * N_H;             // [8192]

    gemv_lif_kernel<<<dim3(N_H / 16), dim3(256), 0, stream>>>(
        W, inp, mem, spk_trace, prediction, neg_trace,
        spk_out, st_out, g_pos, g_neg);

    loss_coef_kernel<<<dim3(1), dim3(256), 0, stream>>>(
        st_out, prediction, neg_trace, g_pos, g_neg, scalar_out, coef);

    outer_kernel<<<dim3(N_H), dim3(256), 0, stream>>>(
        inp, inp_trace, coef, dW);
}